// SimVQ1D_52252572123401
// MI455X (gfx1250) — compile-verified
//
#include <hip/hip_runtime.h>

typedef float v2f __attribute__((ext_vector_type(2)));
typedef float v8f __attribute__((ext_vector_type(8)));

#define NTOK   65536
#define DIM    64
#define KCB    16384
#define NCHUNK 32
#define CHUNKSZ 512
#define NTOT   (NTOK * DIM)   // 4194304
#define MAXTILES 16416        // 262144/16 + 32 slack for per-chunk ceil

// ---- workspace byte offsets ----
#define OFF_CB      0u          // 16384*64 f32      = 4194304 B
#define OFF_CNORM   4194304u    // 16384 f32         = 65536 B
#define OFF_MEANS   4259840u    // 32*64 f32         = 8192 B
#define OFF_M2      4268032u    // 32 f32            = 128 B
#define OFF_TK      4268160u    // 65536*4 u32       = 1048576 B
#define OFF_BEST    5316736u    // 65536 u64         = 524288 B
#define OFF_BINCNT  5841024u    // 32 u32            = 128 B
#define OFF_BINOFF  5841152u    // 33 u32 (pad)      = 256 B
#define OFF_TILEOFF 5841408u    // 33 u32 (pad)      = 256 B
#define OFF_CURSOR  5841664u    // 32 u32            = 128 B
#define OFF_BINNED  5841792u    // 262144 u32        = 1048576 B
#define OFF_COUNTS  6890368u    // 16384 u32         = 65536 B
#define OFF_PARTIAL 6955904u    // 256 f32           = 1024 B

__device__ __forceinline__ unsigned float_key(float f) {
  unsigned u = __float_as_uint(f);
  return (u & 0x80000000u) ? ~u : (u | 0x80000000u);
}

// ---------- init scratch state ----------
__global__ void k_init(unsigned long long* best, unsigned* counts, unsigned* bincnt) {
  int i = blockIdx.x * blockDim.x + threadIdx.x;
  if (i < NTOK)   best[i]   = ~0ull;
  if (i < KCB)    counts[i] = 0u;
  if (i < NCHUNK) bincnt[i] = 0u;
}

// ---------- codebook = embed @ W + bias (WMMA f32 16x16x4) ----------
__global__ void k_proj(const float* __restrict__ embed, const float* __restrict__ W,
                       const float* __restrict__ bias, float* __restrict__ cb) {
  int bm = blockIdx.x, bn = blockIdx.y;
  int lane = threadIdx.x;
  int half = lane >> 4;          // 0: K=k0,k0+1   1: K=k0+2,k0+3
  int lr   = lane & 15;          // row (A) / col (B,C,D)
  v8f acc = {};
  const float* arow = embed + (size_t)(bm * 16 + lr) * DIM;
  for (int k0 = 0; k0 < DIM; k0 += 4) {
    int kk = k0 + half * 2;
    float2 av = *(const float2*)(arow + kk);
    v2f a; a.x = av.x; a.y = av.y;
    v2f b;
    b.x = W[kk * DIM + bn * 16 + lr];
    b.y = W[(kk + 1) * DIM + bn * 16 + lr];
    acc = __builtin_amdgcn_wmma_f32_16x16x4_f32(false, a, false, b, (short)0, acc, false, false);
  }
  float bv = bias[bn * 16 + lr];
#pragma unroll
  for (int r = 0; r < 8; ++r) {
    int row = bm * 16 + r + half * 8;
    cb[(size_t)row * DIM + bn * 16 + lr] = acc[r] + bv;
  }
}

// ---------- per-code squared norms ----------
__global__ void k_cnorm(const float* __restrict__ cb, float* __restrict__ cnorm) {
  int row = blockIdx.x, lane = threadIdx.x;
  float2 v = *(const float2*)(cb + (size_t)row * DIM + lane * 2);
  float s = v.x * v.x + v.y * v.y;
#pragma unroll
  for (int off = 1; off < 32; off <<= 1) s += __shfl_xor(s, off, 32);
  if (lane == 0) cnorm[row] = s;
}

// ---------- chunk means + m2 ----------
__global__ void k_means(const float* __restrict__ cb, float* __restrict__ means,
                        float* __restrict__ m2) {
  __shared__ float red[64];
  int c = blockIdx.x, t = threadIdx.x;
  float s = 0.f;
  for (int r = 0; r < CHUNKSZ; ++r) s += cb[(size_t)(c * CHUNKSZ + r) * DIM + t];
  float mean = s * (1.0f / (float)CHUNKSZ);
  means[c * DIM + t] = mean;
  red[t] = mean * mean;
  __syncthreads();
  for (int off = 32; off > 0; off >>= 1) {
    if (t < off) red[t] += red[t + off];
    __syncthreads();
  }
  if (t == 0) m2[c] = red[0];
}

// ---------- coarse: top-4 nearest chunks per token ----------
__global__ void k_coarse(const float* __restrict__ z, const float* __restrict__ means,
                         const float* __restrict__ m2, unsigned* __restrict__ tk) {
  __shared__ float smean[NCHUNK * DIM];
  __shared__ float sm2[NCHUNK];
  int tid = threadIdx.x;
  for (int i = tid; i < NCHUNK * DIM; i += 256) smean[i] = means[i];
  if (tid < NCHUNK) sm2[tid] = m2[tid];
  __syncthreads();

  int t = blockIdx.x * 256 + tid;
  const float4* zr4 = (const float4*)(z + (size_t)t * DIM);
  float zz = 0.f;
#pragma unroll
  for (int q = 0; q < 16; ++q) {
    float4 v = zr4[q];
    zz += v.x * v.x + v.y * v.y + v.z * v.z + v.w * v.w;
  }
  float d0 = 3.4e38f, d1 = 3.4e38f, d2 = 3.4e38f, d3 = 3.4e38f;
  int i0 = 0, i1 = 0, i2 = 0, i3 = 0;
  for (int c = 0; c < NCHUNK; ++c) {
    const float4* m4 = (const float4*)(smean + c * DIM);
    float dot = 0.f;
#pragma unroll
    for (int q = 0; q < 16; ++q) {
      float4 a = zr4[q]; float4 b = m4[q];
      dot += a.x * b.x + a.y * b.y + a.z * b.z + a.w * b.w;
    }
    float d = zz + sm2[c] - 2.0f * dot;
    if (d < d3) {
      d3 = d; i3 = c;
      if (d3 < d2) { float td = d2; d2 = d3; d3 = td; int ti = i2; i2 = i3; i3 = ti;
        if (d2 < d1) { td = d1; d1 = d2; d2 = td; ti = i1; i1 = i2; i2 = ti;
          if (d1 < d0) { td = d0; d0 = d1; d1 = td; ti = i0; i0 = i1; i1 = ti; }
        }
      }
    }
  }
  unsigned* o = tk + (size_t)t * 4;
  o[0] = (unsigned)i0; o[1] = (unsigned)i1; o[2] = (unsigned)i2; o[3] = (unsigned)i3;
}

// ---------- histogram of (token,chunk) entries per chunk ----------
__global__ void k_hist(const unsigned* __restrict__ tk, unsigned* __restrict__ bincnt) {
  int t = blockIdx.x * 256 + threadIdx.x;
#pragma unroll
  for (int j = 0; j < 4; ++j) atomicAdd(&bincnt[tk[(size_t)t * 4 + j]], 1u);
}

// ---------- serial scan over 32 bins ----------
__global__ void k_scan(const unsigned* __restrict__ bincnt, unsigned* __restrict__ binoff,
                       unsigned* __restrict__ tileoff, unsigned* __restrict__ cursor) {
  if (threadIdx.x == 0) {
    unsigned off = 0, toff = 0;
    for (int c = 0; c < NCHUNK; ++c) {
      binoff[c] = off; tileoff[c] = toff; cursor[c] = off;
      unsigned n = bincnt[c];
      off += n; toff += (n + 15) >> 4;
    }
    binoff[NCHUNK] = off; tileoff[NCHUNK] = toff;
  }
}

// ---------- scatter token ids into per-chunk bins ----------
__global__ void k_scatter(const unsigned* __restrict__ tk, unsigned* __restrict__ cursor,
                          unsigned* __restrict__ binned) {
  int t = blockIdx.x * 256 + threadIdx.x;
#pragma unroll
  for (int j = 0; j < 4; ++j) {
    unsigned c = tk[(size_t)t * 4 + j];
    unsigned pos = atomicAdd(&cursor[c], 1u);
    binned[pos] = (unsigned)t;
  }
}

// ---------- fine search: 16-token tile x 512 codes via WMMA ----------
__global__ void k_fine(const float* __restrict__ z, const float* __restrict__ cb,
                       const float* __restrict__ cnorm, const unsigned* __restrict__ binoff,
                       const unsigned* __restrict__ tileoff, const unsigned* __restrict__ binned,
                       unsigned long long* __restrict__ best) {
  __shared__ unsigned stid[16];
  unsigned b = blockIdx.x;
  int lane = threadIdx.x;
  unsigned ntile = tileoff[NCHUNK];
  if (b >= ntile) return;
  int c = 0;
  while (c < NCHUNK - 1 && tileoff[c + 1] <= b) c++;
  unsigned base = binoff[c] + (b - tileoff[c]) * 16;
  unsigned lim  = binoff[c + 1];
  if (lane < 16) stid[lane] = (base + lane < lim) ? binned[base + lane] : 0xFFFFFFFFu;
  __syncthreads();

  int half = lane >> 4;
  int lr   = lane & 15;
  unsigned mytid = stid[lr];
  const float* zrow = z + (size_t)((mytid == 0xFFFFFFFFu) ? 0u : mytid) * DIM;
  int cbase = c * CHUNKSZ;

  unsigned long long best8[8];
#pragma unroll
  for (int r = 0; r < 8; ++r) best8[r] = ~0ull;

  for (int nb = 0; nb < CHUNKSZ / 16; ++nb) {
    int col = cbase + nb * 16 + lr;
    const float* crow = cb + (size_t)col * DIM;
    if (nb + 1 < CHUNKSZ / 16)
      __builtin_prefetch(cb + (size_t)(col + 16) * DIM, 0, 1);  // global_prefetch_b8
    v8f acc = {};
#pragma unroll
    for (int k0 = 0; k0 < DIM; k0 += 4) {
      int kk = k0 + half * 2;
      float2 av = *(const float2*)(zrow + kk);
      float2 bv = *(const float2*)(crow + kk);
      v2f a; a.x = av.x; a.y = av.y;
      v2f bb; bb.x = bv.x; bb.y = bv.y;
      acc = __builtin_amdgcn_wmma_f32_16x16x4_f32(false, a, false, bb, (short)0, acc, false, false);
    }
    float cn = cnorm[col];
#pragma unroll
    for (int r = 0; r < 8; ++r) {
      float d = cn - 2.0f * acc[r];
      unsigned long long p = ((unsigned long long)float_key(d) << 32) | (unsigned)col;
      best8[r] = (p < best8[r]) ? p : best8[r];
    }
  }
  // reduce over the 16 columns held in each half-wave, then global atomicMin
#pragma unroll
  for (int r = 0; r < 8; ++r) {
    unsigned long long v = best8[r];
#pragma unroll
    for (int off = 1; off < 16; off <<= 1) {
      unsigned long long o = __shfl_xor(v, off, 32);
      v = (o < v) ? o : v;
    }
    if (lr == 0) {
      int row = r + half * 8;
      unsigned t2 = stid[row];
      if (t2 != 0xFFFFFFFFu) atomicMin(&best[t2], v);
    }
  }
}

// ---------- gather quant rows, MSE partials, histogram ----------
__global__ void k_finalize(const float* __restrict__ z, const float* __restrict__ cb,
                           const unsigned long long* __restrict__ best,
                           float* __restrict__ out, unsigned* __restrict__ counts,
                           float* __restrict__ partial) {
  __shared__ float sred[256];
  int t = blockIdx.x * 256 + threadIdx.x;
  unsigned idx = (unsigned)(best[t] & 0xFFFFFFFFull) & (KCB - 1);
  const float* q  = cb + (size_t)idx * DIM;
  const float* zr = z + (size_t)t * DIM;
  float* o = out + (size_t)t * DIM;
  float s = 0.f;
#pragma unroll 8
  for (int k = 0; k < DIM; ++k) {
    float qq = q[k];
    float d = qq - zr[k];
    s += d * d;
    o[k] = qq;
  }
  atomicAdd(&counts[idx], 1u);
  sred[threadIdx.x] = s;
  __syncthreads();
  for (int off = 128; off > 0; off >>= 1) {
    if (threadIdx.x < off) sred[threadIdx.x] += sred[threadIdx.x + off];
    __syncthreads();
  }
  if (threadIdx.x == 0) partial[blockIdx.x] = sred[0];
}

// ---------- scalars: loss, perplexity, usage ----------
__global__ void k_stats(const float* __restrict__ partial, const unsigned* __restrict__ counts,
                        float* __restrict__ out) {
  __shared__ float sh[256];
  int t = threadIdx.x;
  // deterministic tree-sum of 256 block partials
  sh[t] = partial[t];
  __syncthreads();
  for (int off = 128; off > 0; off >>= 1) {
    if (t < off) sh[t] += sh[t + off];
    __syncthreads();
  }
  float sse = sh[0];
  __syncthreads();
  float ent = 0.f, used = 0.f;
  for (int i = t; i < KCB; i += 256) {
    unsigned cnt = counts[i];
    if (cnt > 0u) {
      float p = (float)cnt * (1.0f / (float)NTOK);
      ent += p * logf(p + 1e-10f);
      used += 1.0f;
    }
  }
  sh[t] = ent;
  __syncthreads();
  for (int off = 128; off > 0; off >>= 1) {
    if (t < off) sh[t] += sh[t + off];
    __syncthreads();
  }
  float entsum = sh[0];
  __syncthreads();
  sh[t] = used;
  __syncthreads();
  for (int off = 128; off > 0; off >>= 1) {
    if (t < off) sh[t] += sh[t + off];
    __syncthreads();
  }
  if (t == 0) {
    out[NTOT + 0] = 1.25f * sse * (1.0f / (float)NTOT);  // commit + codebook loss
    out[NTOT + 1] = expf(-entsum);                       // perplexity
    out[NTOT + 2] = sh[0] * (1.0f / (float)KCB);         // usage ratio
  }
}

extern "C" void kernel_launch(void* const* d_in, const int* in_sizes, int n_in,
                              void* d_out, int out_size, void* d_ws, size_t ws_size,
                              hipStream_t stream) {
  const float* z     = (const float*)d_in[0];  // (16,4096,64)
  const float* embed = (const float*)d_in[1];  // (16384,64)
  const float* W     = (const float*)d_in[2];  // (64,64)
  const float* bias  = (const float*)d_in[3];  // (64,)
  float* out = (float*)d_out;

  char* ws = (char*)d_ws;
  float*              cb      = (float*)(ws + OFF_CB);
  float*              cnorm   = (float*)(ws + OFF_CNORM);
  float*              means   = (float*)(ws + OFF_MEANS);
  float*              m2      = (float*)(ws + OFF_M2);
  unsigned*           tk      = (unsigned*)(ws + OFF_TK);
  unsigned long long* best    = (unsigned long long*)(ws + OFF_BEST);
  unsigned*           bincnt  = (unsigned*)(ws + OFF_BINCNT);
  unsigned*           binoff  = (unsigned*)(ws + OFF_BINOFF);
  unsigned*           tileoff = (unsigned*)(ws + OFF_TILEOFF);
  unsigned*           cursor  = (unsigned*)(ws + OFF_CURSOR);
  unsigned*           binned  = (unsigned*)(ws + OFF_BINNED);
  unsigned*           counts  = (unsigned*)(ws + OFF_COUNTS);
  float*              partial = (float*)(ws + OFF_PARTIAL);

  k_init    <<<256, 256, 0, stream>>>(best, counts, bincnt);
  k_proj    <<<dim3(KCB / 16, DIM / 16), 32, 0, stream>>>(embed, W, bias, cb);
  k_cnorm   <<<KCB, 32, 0, stream>>>(cb, cnorm);
  k_means   <<<NCHUNK, 64, 0, stream>>>(cb, means, m2);
  k_coarse  <<<NTOK / 256, 256, 0, stream>>>(z, means, m2, tk);
  k_hist    <<<NTOK / 256, 256, 0, stream>>>(tk, bincnt);
  k_scan    <<<1, 32, 0, stream>>>(bincnt, binoff, tileoff, cursor);
  k_scatter <<<NTOK / 256, 256, 0, stream>>>(tk, cursor, binned);
  k_fine    <<<MAXTILES, 32, 0, stream>>>(z, cb, cnorm, binoff, tileoff, binned, best);
  k_finalize<<<NTOK / 256, 256, 0, stream>>>(z, cb, best, out, counts, partial);
  k_stats   <<<1, 256, 0, stream>>>(partial, counts, out);
}